// HunyuanTopKGate_58076547777302
// MI455X (gfx1250) — compile-verified
//
#include <hip/hip_runtime.h>
#include <hip/hip_bf16.h>

// -------- problem constants (setup_inputs: T=4096, E=64, topk=8, n_group=8, topk_group=4)
#define T_TOK   4096
#define NEXP    64
#define TOPK    8
#define NGRP    8
#define KGRP    4
#define GSZ     8          // experts per group
#define CAP     512        // expert_capacity = max(8, 8*4096/64)
#define TEC     ((size_t)T_TOK * NEXP * CAP)   // 134,217,728

typedef float  v2f __attribute__((ext_vector_type(2)));
typedef float  v4f __attribute__((ext_vector_type(4)));
typedef float  v8f __attribute__((ext_vector_type(8)));

// ---------------------------------------------------------------------------
// 1) Zero-fill the entire output (dominant cost: ~1.07 GB). Non-temporal B128
//    stores so we do not thrash L2 with zeros; real values are scattered later.
// ---------------------------------------------------------------------------
__global__ __launch_bounds__(256) void fill_zero_kernel(float* __restrict__ out, size_t n) {
    size_t n4 = n >> 2;
    size_t i = (size_t)blockIdx.x * blockDim.x + threadIdx.x;
    size_t stride = (size_t)gridDim.x * blockDim.x;
    v4f z = {0.f, 0.f, 0.f, 0.f};
    v4f* out4 = (v4f*)out;
    for (; i < n4; i += stride)
        __builtin_nontemporal_store(z, out4 + i);
    size_t tail = n & 3;
    if (blockIdx.x == 0 && threadIdx.x < tail)
        out[n4 * 4 + threadIdx.x] = 0.f;
}

// ---------------------------------------------------------------------------
// 2) Zero small workspace metadata (int histograms + valid counter).
// ---------------------------------------------------------------------------
__global__ void meta_zero_kernel(int* __restrict__ cnt, int* __restrict__ validCount) {
    for (int i = threadIdx.x; i < TOPK * NEXP; i += blockDim.x) cnt[i] = 0;
    if (threadIdx.x == 0) *validCount = 0;
}

// ---------------------------------------------------------------------------
// 3) Gating kernel: one wave32 per token (8 tokens / 256-thread block).
//    Each lane owns experts {lane, lane+32}. Produces:
//      sel_idx[k][t], sel_rp[k][t]  (slot-major selections + renorm probs)
//      cnt[k][e]                    (per-slot expert histogram, int atomics)
//      gate_partial[block][e]       (masked-gate column sums via WMMA f32 16x16x4)
// ---------------------------------------------------------------------------
__global__ __launch_bounds__(256) void gate_kernel(const float* __restrict__ logits,
                                                   int*   __restrict__ sel_idx,
                                                   float* __restrict__ sel_rp,
                                                   int*   __restrict__ cnt,
                                                   float* __restrict__ gate_partial) {
    const int tid  = threadIdx.x;
    const int lane = tid & 31;
    const int wave = tid >> 5;                 // 0..7 = token within block
    const int tok  = blockIdx.x * 8 + wave;

    __shared__ float lds_g[8][NEXP];           // masked gates, [token-in-block][expert]
    __shared__ float sums_l[2][NEXP];          // per-quad WMMA column sums

    // ---- softmax over 64 logits (2 per lane) ----
    float g0 = logits[(size_t)tok * NEXP + lane];
    float g1 = logits[(size_t)tok * NEXP + 32 + lane];
    float m = fmaxf(g0, g1);
    for (int o = 16; o; o >>= 1) m = fmaxf(m, __shfl_xor(m, o, 32));
    float e0 = expf(g0 - m), e1 = expf(g1 - m);
    float s = e0 + e1;
    for (int o = 16; o; o >>= 1) s += __shfl_xor(s, o, 32);
    float inv = 1.0f / s;
    float p0 = e0 * inv, p1 = e1 * inv;        // softmax gates

    // ---- group scores: max within aligned groups of 8 lanes ----
    float q0 = p0, q1 = p1;
    for (int o = 1; o < 8; o <<= 1) {
        q0 = fmaxf(q0, __shfl_xor(q0, o, 32));
        q1 = fmaxf(q1, __shfl_xor(q1, o, 32));
    }
    float sc[NGRP];
    for (int j = 0; j < 4; ++j) {
        sc[j]     = __shfl(q0, j * 8, 32);     // groups 0..3  (experts 0..31)
        sc[4 + j] = __shfl(q1, j * 8, 32);     // groups 4..7  (experts 32..63)
    }
    // ---- top-4 groups, ties -> lower index (rank-count formulation) ----
    unsigned gm = 0;
    for (int i = 0; i < NGRP; ++i) {
        int beats = 0;
        for (int j = 0; j < NGRP; ++j)
            beats += (sc[j] > sc[i]) || (sc[j] == sc[i] && j < i);
        if (beats < KGRP) gm |= 1u << i;
    }
    float mg0 = ((gm >> (lane >> 3)) & 1)        ? p0 : 0.f;
    float mg1 = ((gm >> (4 + (lane >> 3))) & 1)  ? p1 : 0.f;
    lds_g[wave][lane]      = mg0;
    lds_g[wave][lane + 32] = mg1;

    // ---- top-8 experts among masked gates (iterative wave argmax, ties -> lower idx) ----
    float v0 = mg0, v1 = mg1;
    float selv[TOPK]; int seli[TOPK]; float ssum = 0.f;
    for (int k = 0; k < TOPK; ++k) {
        float cv; int ci;
        if (v0 >= v1) { cv = v0; ci = lane; } else { cv = v1; ci = lane + 32; }
        for (int o = 16; o; o >>= 1) {
            float ov = __shfl_xor(cv, o, 32);
            int   oi = __shfl_xor(ci, o, 32);
            if (ov > cv || (ov == cv && oi < ci)) { cv = ov; ci = oi; }
        }
        selv[k] = cv; seli[k] = ci; ssum += cv;   // wave-uniform after reduce
        if (ci == lane)      v0 = -1.f;
        if (ci == lane + 32) v1 = -1.f;
    }
    float gs  = fmaxf(ssum, 1.1920929e-07f);      // FP32_EPS clamp (norm_topk_prob)
    float rcp = 1.0f / gs;
    if (lane < TOPK) {
        int k = lane;
        sel_idx[k * T_TOK + tok] = seli[k];
        sel_rp [k * T_TOK + tok] = selv[k] * rcp;
        atomicAdd(&cnt[k * NEXP + seli[k]], 1);   // int atomic: deterministic
    }
    __syncthreads();

    // ---- WMMA f32 16x16x4: column sums of masked gates over 4 tokens ----
    // wave w handles quad = w>>2 (tokens quad*4..+3), tile = w&3 (experts tile*16..+15).
    // A = ones(16x4); B[k][n] = lds_g[quad*4+k][tile*16+n]; D row0 = column sums.
    // 32-bit A/B layout: VGPR0 holds K={0,2} (lanes 0-15 / 16-31), VGPR1 K={1,3}.
    {
        int quad = wave >> 2, tile = wave & 3;
        int l  = lane & 15;
        int hi = lane >> 4;
        int tokBase = quad * 4;
        v2f a; a.x = 1.f; a.y = 1.f;
        v2f b;
        b.x = lds_g[tokBase + (hi ? 2 : 0)][tile * 16 + l];
        b.y = lds_g[tokBase + (hi ? 3 : 1)][tile * 16 + l];
        v8f c = {};
        c = __builtin_amdgcn_wmma_f32_16x16x4_f32(
                /*neg_a=*/false, a, /*neg_b=*/false, b,
                /*c_mod=*/(short)0, c, /*reuse_a=*/false, /*reuse_b=*/false);
        // D VGPR0: lanes 0-15 -> (M=0, N=lane) = column sum for expert tile*16+l
        if (hi == 0) sums_l[quad][tile * 16 + l] = c[0];
    }
    __syncthreads();
    if (tid < NEXP)
        gate_partial[(size_t)blockIdx.x * NEXP + tid] = sums_l[0][tid] + sums_l[1][tid];
}

// ---------------------------------------------------------------------------
// 4) Slot-major exclusive offsets per expert + exp_counts output (as float).
// ---------------------------------------------------------------------------
__global__ void offsets_kernel(const int* __restrict__ cnt, int* __restrict__ off,
                               float* __restrict__ out_counts) {
    int e = threadIdx.x;               // 0..63
    int running = 0;
    for (int k = 0; k < TOPK; ++k) {
        off[k * NEXP + e] = running;
        running += cnt[k * NEXP + e];
    }
    out_counts[e] = (float)running;    // exp_counts (uncapped totals)
}

// ---------------------------------------------------------------------------
// 5) Scatter: one wave per (slot, expert). Stable rank via ballot prefix over
//    tokens in order => matches the reference's slot-major cumsum exactly.
// ---------------------------------------------------------------------------
__global__ __launch_bounds__(32) void scatter_kernel(const int*   __restrict__ sel_idx,
                                                     const float* __restrict__ sel_rp,
                                                     const int*   __restrict__ off,
                                                     float* __restrict__ out_cw,
                                                     float* __restrict__ out_dm,
                                                     int*   __restrict__ validCount) {
    int pair = blockIdx.x;             // 0..511
    int k = pair >> 6, e = pair & 63;
    int lane = threadIdx.x;
    int base = off[pair];
    int myvalid = 0;
    for (int t0 = 0; t0 < T_TOK; t0 += 32) {
        int t = t0 + lane;
        int idx = sel_idx[k * T_TOK + t];
        bool match = (idx == e);
        unsigned mask = (unsigned)__ballot(match);      // wave32: low 32 bits
        int prefix = __popc(mask & ((1u << lane) - 1u));
        if (match) {
            int rank = base + prefix;
            if (rank < CAP) {
                float rp = sel_rp[k * T_TOK + t];
                size_t o = (size_t)t * (NEXP * CAP) + (size_t)e * CAP + rank;
                out_cw[o] = rp;
                out_dm[o] = 1.0f;
                ++myvalid;
            }
        }
        base += __popc(mask);
    }
    for (int o = 16; o; o >>= 1) myvalid += __shfl_xor(myvalid, o, 32);
    if (lane == 0 && myvalid) atomicAdd(validCount, myvalid);
}

// ---------------------------------------------------------------------------
// 6) Finalize: l_aux = E * sum_e(count[e]*gateSum[e]) / T^2 ; capacity rate.
//    Fixed-order summation => deterministic.
// ---------------------------------------------------------------------------
__global__ void finalize_kernel(const float* __restrict__ gate_partial,
                                const int*   __restrict__ cnt,
                                const int*   __restrict__ validCount,
                                float* __restrict__ d_out) {
    __shared__ float prods[NEXP];
    int e = threadIdx.x;               // 0..63
    float s = 0.f;
    for (int b = 0; b < T_TOK / 8; ++b)           // 512 blocks, fixed order
        s += gate_partial[(size_t)b * NEXP + e];
    int tot = 0;
    for (int k = 0; k < TOPK; ++k) tot += cnt[k * NEXP + e];
    prods[e] = (float)tot * s;
    __syncthreads();
    if (e == 0) {
        float acc = 0.f;
        for (int i = 0; i < NEXP; ++i) acc += prods[i];   // fixed order
        d_out[0] = (float)NEXP * acc / ((float)T_TOK * (float)T_TOK);
        d_out[1] = (float)(*validCount) / (float)(T_TOK * TOPK);
    }
}

// ---------------------------------------------------------------------------
extern "C" void kernel_launch(void* const* d_in, const int* in_sizes, int n_in,
                              void* d_out, int out_size, void* d_ws, size_t ws_size,
                              hipStream_t stream) {
    const float* logits = (const float*)d_in[0];
    float* out = (float*)d_out;

    // workspace layout (bytes)
    char* ws = (char*)d_ws;
    int*   sel_idx      = (int*)  (ws);                       // 8*4096 ints  = 128 KB
    float* sel_rp       = (float*)(ws + 131072);              // 8*4096 f32   = 128 KB
    int*   cnt          = (int*)  (ws + 262144);              // 512 ints
    int*   off          = (int*)  (ws + 264192);              // 512 ints
    float* gate_partial = (float*)(ws + 266240);              // 512*64 f32   = 128 KB
    int*   validCount   = (int*)  (ws + 397312);              // 1 int

    float* out_cw     = out + 2;                  // combine_weights [T,E,C]
    float* out_dm     = out + 2 + TEC;            // dispatch_mask   [T,E,C] (as float)
    float* out_counts = out + 2 + 2 * TEC;        // exp_counts [E]  (as float)

    // total output = 2 + 2*TEC + 64 floats (~1.07 GB) -> NT zero fill first
    fill_zero_kernel<<<4096, 256, 0, stream>>>(out, (size_t)out_size);
    meta_zero_kernel<<<1, 256, 0, stream>>>(cnt, validCount);
    gate_kernel<<<T_TOK / 8, 256, 0, stream>>>(logits, sel_idx, sel_rp, cnt, gate_partial);
    offsets_kernel<<<1, NEXP, 0, stream>>>(cnt, off, out_counts);
    scatter_kernel<<<TOPK * NEXP, 32, 0, stream>>>(sel_idx, sel_rp, off, out_cw, out_dm, validCount);
    finalize_kernel<<<1, NEXP, 0, stream>>>(gate_partial, cnt, validCount, out);
}